// EMDLoss_69526930588392
// MI455X (gfx1250) — compile-verified
//
#include <hip/hip_runtime.h>
#include <math.h>

// EMD (Sinkhorn) for N=M=1024, D=3, EPS=0.01.
// Phase 0: cost matrix C (and C^T) via V_WMMA_F32_16X16X4_F32 (K=3 padded to 4).
// Phase 1: persistent kernel, Sinkhorn iterations in log domain.
//          - f/g vectors staged block-locally into LDS with
//            GLOBAL_LOAD_ASYNC_TO_LDS_B128 (+ s_wait_asynccnt)
//          - per-row residuals register-cached: single global read per element
//          - in-kernel grid barrier via agent-scope acq/rel atomics
// Phase 2: deterministic <P, C> reduction.
//
// Workspace (floats): C[1M] | CT[1M] | f[1024] | g[1024] | partial[64] | bar[2]
// ~8.4 MB total -> resident in MI455X's 192 MB L2; HBM is touched once.

#define N_PTS 1024
#define NB 64          // persistent blocks
#define TPB 256        // 8 waves of 32
#define EPS_R 0.01f
#define INV_EPS 100.0f
#define LOG_A (-6.93147180559945f)      // -log(1024)

typedef float v2f __attribute__((ext_vector_type(2)));
typedef float v8f __attribute__((ext_vector_type(8)));

// ---------------- grid barrier (sense via generation counter) ----------------
__device__ __forceinline__ void grid_barrier(unsigned* cnt, unsigned* gen) {
  __syncthreads();
  if (threadIdx.x == 0) {
    unsigned my_gen = __hip_atomic_load(gen, __ATOMIC_ACQUIRE, __HIP_MEMORY_SCOPE_AGENT);
    unsigned arrived = __hip_atomic_fetch_add(cnt, 1u, __ATOMIC_ACQ_REL, __HIP_MEMORY_SCOPE_AGENT);
    if (arrived == NB - 1) {
      __hip_atomic_store(cnt, 0u, __ATOMIC_RELAXED, __HIP_MEMORY_SCOPE_AGENT);
      __hip_atomic_fetch_add(gen, 1u, __ATOMIC_RELEASE, __HIP_MEMORY_SCOPE_AGENT);
    } else {
      while (__hip_atomic_load(gen, __ATOMIC_ACQUIRE, __HIP_MEMORY_SCOPE_AGENT) == my_gen) {
        __builtin_amdgcn_s_sleep(2);
      }
    }
  }
  __syncthreads();
}

// ---- async stage: copy 4KB vector (1024 f32) global -> LDS offset 0 --------
// One GLOBAL_LOAD_ASYNC_TO_LDS_B128 per thread (GVS mode: saddr base + vaddr).
// LDS dest VGPR holds byte offset within the wave's LDS allocation; our single
// __shared__ array sits at group-segment offset 0.
__device__ __forceinline__ void stage_vec_async(const float* __restrict__ src) {
  unsigned voff = (unsigned)threadIdx.x * 16u;   // 256 threads x 16B = 4KB
  asm volatile("global_load_async_to_lds_b128 %0, %1, %2"
               :
               : "v"(voff), "v"(voff), "s"(src)
               : "memory");
  asm volatile("s_wait_asynccnt 0x0" ::: "memory");
  __syncthreads();
}

// ---------------- phase init: zero f, g, partials, barrier ----------------
__global__ void emd_init_kernel(float* f, float* g, float* partial, unsigned* bar) {
  int t = threadIdx.x;
  for (int i = t; i < N_PTS; i += TPB) { f[i] = 0.0f; g[i] = 0.0f; }
  if (t < NB) partial[t] = 0.0f;
  if (t < 2) bar[t] = 0u;
}

// ---------------- phase 0: cost matrix via WMMA f32 16x16x4 ----------------
// One wave per 16x16 tile of C. A = target tile (16x4, K: x,y,z,0),
// B = output tile as KxN (N on lanes, K on vgpr+lanehalf).
__global__ __launch_bounds__(TPB) void emd_cost_kernel(
    const float* __restrict__ tgt, const float* __restrict__ outp,
    float* __restrict__ C, float* __restrict__ CT) {
  const int wave = (blockIdx.x * TPB + threadIdx.x) >> 5;  // 0..4095
  const int lane = threadIdx.x & 31;
  const int half = lane >> 4;       // 0: K=0,1   1: K=2,3(pad)
  const int l16  = lane & 15;
  const int tI = wave >> 6;         // row tile 0..63
  const int tJ = wave & 63;         // col tile 0..63

  const int mi = tI * 16 + l16;
  const float txA = tgt[mi * 3 + 0];
  const float tyA = tgt[mi * 3 + 1];
  const float tzA = tgt[mi * 3 + 2];
  v2f a;
  a.x = half ? tzA : txA;
  a.y = half ? 0.0f : tyA;

  const int nj = tJ * 16 + l16;
  const float ox = outp[nj * 3 + 0];
  const float oy = outp[nj * 3 + 1];
  const float oz = outp[nj * 3 + 2];
  v2f b;
  b.x = half ? oz : ox;
  b.y = half ? 0.0f : oy;

  v8f acc = {};
  // D[r][lane] = dot(t_M, o_N), M = r + 8*half, N = l16
  acc = __builtin_amdgcn_wmma_f32_16x16x4_f32(
      false, a, false, b, (short)0, acc, false, false);

  const float on = ox * ox + oy * oy + oz * oz;
  const int gj = nj;
#pragma unroll
  for (int r = 0; r < 8; ++r) {
    const int gi = tI * 16 + 8 * half + r;
    const float tx = tgt[gi * 3 + 0];    // broadcast loads
    const float ty = tgt[gi * 3 + 1];
    const float tz = tgt[gi * 3 + 2];
    const float tn = tx * tx + ty * ty + tz * tz;
    const float c  = tn + on - 2.0f * acc[r];
    C [gi * N_PTS + gj] = c;
    CT[gj * N_PTS + gi] = c;
  }
}

// ---------------- phase 1+2: persistent Sinkhorn + objective ----------------
__global__ __launch_bounds__(TPB) void emd_sinkhorn_kernel(
    const float* __restrict__ C, const float* __restrict__ CT,
    float* __restrict__ f, float* __restrict__ g,
    unsigned* bar, float* partial, float* out, const int* iters_p) {
  __shared__ float smem[N_PTS + 8];  // [0,1024): staged vector; [1024,1032): wave sums
  float* vec  = smem;
  float* wsum = smem + N_PTS;

  const int lane  = threadIdx.x & 31;
  const int wid   = threadIdx.x >> 5;              // 0..7
  const int gwave = blockIdx.x * 8 + wid;          // 0..511
  const int niter = iters_p[0];
  unsigned* cnt = bar;
  unsigned* gen = bar + 1;

  for (int it = 0; it < niter; ++it) {
    // ---- f update (rows of C, uses g) ----
    stage_vec_async(g);
#pragma unroll
    for (int rr = 0; rr < 2; ++rr) {
      const int i = gwave * 2 + rr;
      const float* __restrict__ row = C + i * N_PTS;
      float x[32];
      float m = -__builtin_inff();
#pragma unroll
      for (int u = 0; u < 32; ++u) {
        const int j = lane + u * 32;
        x[u] = vec[j] - row[j];          // single global read per element
        m = fmaxf(m, x[u]);
      }
      for (int off = 16; off > 0; off >>= 1)
        m = fmaxf(m, __shfl_xor(m, off, 32));
      float s = 0.0f;
#pragma unroll
      for (int u = 0; u < 32; ++u)
        s += __expf((x[u] - m) * INV_EPS);
      for (int off = 16; off > 0; off >>= 1)
        s += __shfl_xor(s, off, 32);
      if (lane == 0) f[i] = EPS_R * LOG_A - m - EPS_R * __logf(s);
    }
    grid_barrier(cnt, gen);

    // ---- g update (rows of CT = columns of C, uses f) ----
    stage_vec_async(f);
#pragma unroll
    for (int rr = 0; rr < 2; ++rr) {
      const int j = gwave * 2 + rr;
      const float* __restrict__ col = CT + j * N_PTS;
      float x[32];
      float m = -__builtin_inff();
#pragma unroll
      for (int u = 0; u < 32; ++u) {
        const int i = lane + u * 32;
        x[u] = vec[i] - col[i];
        m = fmaxf(m, x[u]);
      }
      for (int off = 16; off > 0; off >>= 1)
        m = fmaxf(m, __shfl_xor(m, off, 32));
      float s = 0.0f;
#pragma unroll
      for (int u = 0; u < 32; ++u)
        s += __expf((x[u] - m) * INV_EPS);
      for (int off = 16; off > 0; off >>= 1)
        s += __shfl_xor(s, off, 32);
      if (lane == 0) g[j] = EPS_R * LOG_A - m - EPS_R * __logf(s);
    }
    grid_barrier(cnt, gen);
  }

  // ---- objective: sum_ij exp((f_i + g_j - C_ij)/eps) * C_ij ----
  stage_vec_async(g);
  float acc = 0.0f;
#pragma unroll
  for (int rr = 0; rr < 2; ++rr) {
    const int i = gwave * 2 + rr;
    const float fi = f[i];
    const float* __restrict__ row = C + i * N_PTS;
    for (int j = lane; j < N_PTS; j += 32) {
      const float c = row[j];
      acc += __expf((fi + vec[j] - c) * INV_EPS) * c;
    }
  }
  for (int off = 16; off > 0; off >>= 1)
    acc += __shfl_xor(acc, off, 32);
  if (lane == 0) wsum[wid] = acc;
  __syncthreads();
  if (threadIdx.x == 0) {
    float b = 0.0f;
#pragma unroll
    for (int w = 0; w < 8; ++w) b += wsum[w];
    partial[blockIdx.x] = b;
  }
  grid_barrier(cnt, gen);
  if (blockIdx.x == 0 && threadIdx.x == 0) {
    float total = 0.0f;
    for (int b2 = 0; b2 < NB; ++b2) total += partial[b2];  // fixed order
    out[0] = total;
  }
}

extern "C" void kernel_launch(void* const* d_in, const int* in_sizes, int n_in,
                              void* d_out, int out_size, void* d_ws, size_t ws_size,
                              hipStream_t stream) {
  const float* tgt  = (const float*)d_in[0];   // [1024,3]
  const float* outp = (const float*)d_in[1];   // [1024,3]
  const int*   itp  = (const int*)d_in[2];     // scalar iters

  float* ws = (float*)d_ws;
  float* C       = ws;                                   // 1M floats
  float* CT      = ws + (size_t)N_PTS * N_PTS;           // 1M floats
  float* f       = CT + (size_t)N_PTS * N_PTS;           // 1024
  float* g       = f + N_PTS;                            // 1024
  float* partial = g + N_PTS;                            // 64
  unsigned* bar  = (unsigned*)(partial + NB);            // 2 u32

  float* outv = (float*)d_out;

  emd_init_kernel<<<1, TPB, 0, stream>>>(f, g, partial, bar);
  emd_cost_kernel<<<(N_PTS / 16) * (N_PTS / 16) / 8, TPB, 0, stream>>>(tgt, outp, C, CT);
  emd_sinkhorn_kernel<<<NB, TPB, 0, stream>>>(C, CT, f, g, bar, partial, outv, itp);
}